// TemporalAttentionLayer_80796924772962
// MI455X (gfx1250) — compile-verified
//
#include <hip/hip_runtime.h>
#include <hip/hip_bf16.h>
#include <math.h>

// ---------------------------------------------------------------------------
// TemporalAttentionLayer, MI455X (gfx1250, wave32)
//
//   stage 1: single streaming pass over x (1 GiB) -> lsh[b,c,t], M[b,c,t]
//            (memory-bound: 23.3 TB/s floor ~46us; VALU FMAs suffice)
//   stage 2: per-batch fused epilogue in LDS with v_wmma_f32_16x16x4_f32:
//            product = lsh^T @ M (K=64), S = sigmoid(product + b_e),
//            E = V_e @ S (K=128), softmax over axis 1, write out.
// ---------------------------------------------------------------------------

#define Bn   16
#define Nn   2048
#define Cn   64
#define Tn   128
#define NSPLIT 32
#define NCHUNK (Nn / NSPLIT)   // 64

typedef float v2f __attribute__((ext_vector_type(2)));
typedef float v8f __attribute__((ext_vector_type(8)));

// ---------------------------------------------------------------------------
__global__ void ta_zero_ws(float* __restrict__ p, int nelem) {
  int i = blockIdx.x * blockDim.x + threadIdx.x;
  if (i < nelem) p[i] = 0.0f;
}

// ---------------------------------------------------------------------------
// Stage 1: block = (b, n-split), 256 threads (8 waves).
// thread -> t = tid&127, c-range = [32*(tid>>7), +32)
// Per n:  accL[c][t] += U1[n] * x[b,n,c,t]
//         r[t]       = sum_c U3[c] * x[b,n,c,t]      (LDS cross-half combine)
//         accM[c][t] += U2[c,n] * r[t]
// Atomically fold into global accumulators (32 partials per target elem).
// ---------------------------------------------------------------------------
__global__ void __launch_bounds__(256)
ta_stage1(const float* __restrict__ x,  const float* __restrict__ U1,
          const float* __restrict__ U2, const float* __restrict__ U3,
          float* __restrict__ lsh_ws,   float* __restrict__ M_ws) {
  const int b     = blockIdx.x;
  const int split = blockIdx.y;
  const int tid   = threadIdx.x;
  const int t     = tid & (Tn - 1);
  const int chalf = tid >> 7;          // wave-uniform
  const int c0    = chalf * 32;

  __shared__ float rbuf[2][2][Tn];     // double-buffered partial r[t]

  float accL[32], accM[32], u3r[32];
#pragma unroll
  for (int i = 0; i < 32; ++i) { accL[i] = 0.0f; accM[i] = 0.0f; }
#pragma unroll
  for (int i = 0; i < 32; ++i) u3r[i] = U3[c0 + i];   // wave-uniform -> SGPRs

  const int n0 = split * NCHUNK;
  for (int nn = 0; nn < NCHUNK; ++nn) {
    const int n = n0 + nn;
    const float u1n = U1[n];
    const float* xb = x + (((size_t)b * Nn + n) * Cn + c0) * Tn + t;
    __builtin_prefetch(xb + (size_t)Cn * Tn, 0, 0);   // global_prefetch_b8: next n

    float partial = 0.0f;
#pragma unroll
    for (int i = 0; i < 32; ++i) {
      const float v = xb[(size_t)i * Tn];             // 128B coalesced per wave
      accL[i] = fmaf(u1n, v, accL[i]);
      partial = fmaf(u3r[i], v, partial);
    }

    const int buf = nn & 1;
    rbuf[buf][chalf][t] = partial;
    __syncthreads();
    const float rsum = rbuf[buf][0][t] + rbuf[buf][1][t];

    const float* u2col = U2 + n;                      // U2[c][n], wave-uniform idx
#pragma unroll
    for (int i = 0; i < 32; ++i)
      accM[i] = fmaf(u2col[(size_t)(c0 + i) * Nn], rsum, accM[i]);
  }

  float* lp = lsh_ws + ((size_t)b * Cn + c0) * Tn + t;
  float* mp = M_ws   + ((size_t)b * Cn + c0) * Tn + t;
#pragma unroll
  for (int i = 0; i < 32; ++i) {
    unsafeAtomicAdd(&lp[(size_t)i * Tn], accL[i]);    // global_atomic_add_f32
    unsafeAtomicAdd(&mp[(size_t)i * Tn], accM[i]);
  }
}

// ---------------------------------------------------------------------------
// Stage 2: one block per batch, 256 threads = 8 waves, dynamic LDS.
//   LDS strides padded for conflict-free access (64 banks, wave32).
// ---------------------------------------------------------------------------
#define LSTR 130   // lsh / M rows  (64 x 130)
#define SSTR 130   // S rows        (128 x 130)
#define ESTR 129   // E rows        (128 x 129), odd stride -> col reads clean
#define VSTR 129   // V_e rows      (128 x 129)

#define SMEM_S   (Tn * SSTR)                       // 16640 floats
#define SMEM_E   (Tn * ESTR)                       // 16512 floats
#define SMEM_SCR (2 * Cn * LSTR)                   // 16640 floats (>= Tn*VSTR)
#define SMEM_FLOATS (SMEM_S + SMEM_E + SMEM_SCR)   // 49792 -> 199168 bytes

__global__ void __launch_bounds__(256)
ta_stage2(const float* __restrict__ lsh_ws, const float* __restrict__ M_ws,
          const float* __restrict__ be,     const float* __restrict__ Ve,
          float* __restrict__ out) {
  extern __shared__ float smem[];
  float* SL  = smem;                 // S = sigmoid(product + b_e), [s][u]
  float* EL  = SL + SMEM_S;          // E, [t][u]
  float* SCR = EL + SMEM_E;          // phase A: lsh|M ; phase B: V_e
  float* lshL = SCR;                 // [c][t], stride LSTR
  float* ML   = SCR + Cn * LSTR;     // [c][s], stride LSTR
  float* VL   = SCR;                 // [t][s], stride VSTR (reuse after barrier)

  const int b    = blockIdx.x;
  const int tid  = threadIdx.x;
  const int wave = tid >> 5;         // 0..7
  const int lane = tid & 31;
  const int half = lane >> 4;        // 0|1
  const int l16  = lane & 15;
  const int koff = half * 2;         // A/B fragment K sub-offset (ISA 7.12.2)

  // ---- load lsh[b], M[b] into LDS ----
  const float* lg = lsh_ws + (size_t)b * Cn * Tn;
  const float* mg = M_ws   + (size_t)b * Cn * Tn;
  for (int idx = tid; idx < Cn * Tn; idx += 256) {
    const int c = idx >> 7, tt = idx & (Tn - 1);
    lshL[c * LSTR + tt] = lg[idx];
    ML[c * LSTR + tt]   = mg[idx];
  }
  __syncthreads();

  // ---- phase A: product = lsh^T @ M (128x128, K=64), sigmoid+bias -> SL ----
  const int tile_m = wave * 16;
  for (int tn = 0; tn < 8; ++tn) {
    const int tile_n = tn * 16;
    v8f acc = {};
#pragma unroll
    for (int k0 = 0; k0 < Cn; k0 += 4) {
      const int ka = k0 + koff;
      v2f a, bb;
      a.x  = lshL[(ka + 0) * LSTR + tile_m + l16];   // A[M=t][K=c]
      a.y  = lshL[(ka + 1) * LSTR + tile_m + l16];
      bb.x = ML[(ka + 0) * LSTR + tile_n + l16];     // B[K=c][N=s]
      bb.y = ML[(ka + 1) * LSTR + tile_n + l16];
      acc = __builtin_amdgcn_wmma_f32_16x16x4_f32(
                false, a, false, bb, (short)0, acc, false, false);
    }
#pragma unroll
    for (int i = 0; i < 8; ++i) {
      const int row = tile_m + i + half * 8;
      const int col = tile_n + l16;
      const float v = acc[i] + be[row * Tn + col];
      SL[row * SSTR + col] = 1.0f / (1.0f + __expf(-v));
    }
  }
  __syncthreads();

  // ---- load V_e into LDS (reuses lsh/M space) ----
  for (int idx = tid; idx < Tn * Tn; idx += 256) {
    const int r = idx >> 7, s = idx & (Tn - 1);
    VL[r * VSTR + s] = Ve[idx];
  }
  __syncthreads();

  // ---- phase B: E = V_e @ S (128x128, K=128) -> EL ----
  for (int tn = 0; tn < 8; ++tn) {
    const int tile_n = tn * 16;
    v8f acc = {};
#pragma unroll
    for (int k0 = 0; k0 < Tn; k0 += 4) {
      const int ka = k0 + koff;
      v2f a, bb;
      a.x  = VL[(tile_m + l16) * VSTR + ka + 0];     // A[M=t][K=s]
      a.y  = VL[(tile_m + l16) * VSTR + ka + 1];
      bb.x = SL[(ka + 0) * SSTR + tile_n + l16];     // B[K=s][N=u]
      bb.y = SL[(ka + 1) * SSTR + tile_n + l16];
      acc = __builtin_amdgcn_wmma_f32_16x16x4_f32(
                false, a, false, bb, (short)0, acc, false, false);
    }
#pragma unroll
    for (int i = 0; i < 8; ++i) {
      const int row = tile_m + i + half * 8;
      EL[row * ESTR + tile_n + l16] = acc[i];
    }
  }
  __syncthreads();

  // ---- phase C: softmax over t (axis 1), column-wise, odd-stride LDS ----
  if (tid < Tn) {
    const int u = tid;
    float mx = -INFINITY;
    for (int tt = 0; tt < Tn; ++tt) mx = fmaxf(mx, EL[tt * ESTR + u]);
    float sum = 0.0f;
    for (int tt = 0; tt < Tn; ++tt) {
      const float e = __expf(EL[tt * ESTR + u] - mx);
      EL[tt * ESTR + u] = e;
      sum += e;
    }
    const float inv = 1.0f / sum;
    float* ob = out + (size_t)b * Tn * Tn + u;
    for (int tt = 0; tt < Tn; ++tt) ob[tt * Tn] = EL[tt * ESTR + u] * inv;
  }
}

// ---------------------------------------------------------------------------
extern "C" void kernel_launch(void* const* d_in, const int* in_sizes, int n_in,
                              void* d_out, int out_size, void* d_ws, size_t ws_size,
                              hipStream_t stream) {
  const float* x  = (const float*)d_in[0];
  const float* U1 = (const float*)d_in[1];
  const float* U2 = (const float*)d_in[2];
  const float* U3 = (const float*)d_in[3];
  const float* be = (const float*)d_in[4];
  const float* Ve = (const float*)d_in[5];
  float* out = (float*)d_out;

  float* lsh_ws = (float*)d_ws;                 // B*C*T
  float* M_ws   = lsh_ws + (size_t)Bn * Cn * Tn;

  const int accElems = 2 * Bn * Cn * Tn;        // 262144 floats (1 MB)
  ta_zero_ws<<<(accElems + 255) / 256, 256, 0, stream>>>(lsh_ws, accElems);

  dim3 g1(Bn, NSPLIT);
  ta_stage1<<<g1, 256, 0, stream>>>(x, U1, U2, U3, lsh_ws, M_ws);

  const size_t smemBytes = (size_t)SMEM_FLOATS * sizeof(float); // ~199 KB < 320 KB
  ta_stage2<<<Bn, 256, smemBytes, stream>>>(lsh_ws, M_ws, be, Ve, out);
}